// GCN_4440996184348
// MI455X (gfx1250) — compile-verified
//
#include <hip/hip_runtime.h>

// ---------------------------------------------------------------------------
// 2-layer GAT for MI455X (gfx1250, wave32).
//  - Dense h = x @ W uses V_WMMA_F32_16X16X4_F32 (full fp32).
//    A (x rows) is padded to stride 104 so A fragments are unconditional b64
//    loads; W is pre-swizzled into the exact per-lane B-fragment stream
//    (Bpk[step][tile][lane] = {W[kk,col], W[kk+1,col]}, zero-padded), so the
//    inner loop is 8 b64 loads + 7 WMMAs with EXEC all-ones and no branches.
//  - Edge phases are L2-resident (~77 MB < 192 MB L2): u32 atomic max on
//    order-preserving float encoding, global_atomic_add_f32 for sums, and a
//    wave-per-edge / lane-per-feature coalesced scatter.
// ---------------------------------------------------------------------------

#define N_NODES 50000
#define N_EDGES 800000
#define D       101
#define PD      104                   // padded feature stride (mult of 4)
#define EA      (N_EDGES + N_NODES)   // edges + self loops
#define L_LAYERS 2
#define NEG_SLOPE 0.2f

#define ROWTILES (N_NODES / 16)       // 3125 (exact)
#define CTILES   7                    // ceil(101/16)
#define KSTEPS   (PD / 4)             // 26
#define BPK_ELEMS (KSTEPS * CTILES * 32)

typedef __attribute__((ext_vector_type(2))) float v2f;
typedef __attribute__((ext_vector_type(8))) float v8f;

// Order-preserving float -> uint encoding so we can use integer atomic max.
__device__ __forceinline__ unsigned f2ord(float f) {
    unsigned u = __float_as_uint(f);
    return (u & 0x80000000u) ? ~u : (u | 0x80000000u);
}
__device__ __forceinline__ float ord2f(unsigned u) {
    unsigned v = (u & 0x80000000u) ? (u & 0x7FFFFFFFu) : ~u;
    return __uint_as_float(v);
}
#define ORD_NEG_INF 0x007FFFFFu  // f2ord(-inf)

__device__ __forceinline__ void edge_sd(const long long* __restrict__ ei,
                                        int e, int& s, int& d) {
    if (e < N_EDGES) { s = (int)ei[e]; d = (int)ei[N_EDGES + e]; }
    else             { s = d = e - N_EDGES; }          // appended self loop
}

// ---------------------------------------------------------------------------
// Pack X[N,101] -> Xp[N,104] (zero pad).  Layer-0 input only.
// ---------------------------------------------------------------------------
__global__ __launch_bounds__(256)
void gat_pack_x(const float* __restrict__ X, float* __restrict__ Xp) {
    int i = blockIdx.x * blockDim.x + threadIdx.x;
    if (i >= N_NODES * PD) return;
    int n = i / PD, d = i % PD;
    Xp[i] = (d < D) ? X[(size_t)n * D + d] : 0.f;
}

// ---------------------------------------------------------------------------
// Pre-swizzle W[101,101] into the per-lane WMMA B-fragment stream:
//   Bpk[(step*CTILES + c)*32 + lane] = { W[kk,col], W[kk+1,col] }
// with kk = 4*step + 2*(lane>>4), col = 16*c + (lane&15); zero outside [0,D).
// ---------------------------------------------------------------------------
__global__ __launch_bounds__(256)
void gat_pack_b(const float* __restrict__ W, v2f* __restrict__ Bpk) {
    int i = blockIdx.x * blockDim.x + threadIdx.x;
    if (i >= BPK_ELEMS) return;
    int lane = i & 31;
    int c    = (i >> 5) % CTILES;
    int step = i / (CTILES * 32);
    int kk   = step * 4 + 2 * (lane >> 4);
    int col  = c * 16 + (lane & 15);
    v2f b;
    b.x = (kk     < D && col < D) ? W[(size_t)kk * D + col]       : 0.f;
    b.y = (kk + 1 < D && col < D) ? W[(size_t)(kk + 1) * D + col] : 0.f;
    Bpk[i] = b;
}

// ---------------------------------------------------------------------------
// GEMM: Hp[N,104] = Xp[N,104] * W  via fp32 WMMA 16x16x4.
// One wave per 16-row strip computes all 7 col tiles; inner loop is
// 1 A-load(b64) + 7 B-loads(b64) + 7 WMMAs, branch-free, EXEC all-ones.
// Pad cols (101..103) come out exactly 0 because Bpk pads are 0.
// ---------------------------------------------------------------------------
__global__ __launch_bounds__(256)
void gat_gemm_wmma(const float* __restrict__ Xp, const v2f* __restrict__ Bpk,
                   float* __restrict__ Hp) {
    int wave = (int)((blockIdx.x * blockDim.x + threadIdx.x) >> 5);
    if (wave >= ROWTILES) return;          // whole-wave uniform exit
    int lane = threadIdx.x & 31;
    int half = lane >> 4;                  // 0: lanes 0-15, 1: lanes 16-31
    int t    = lane & 15;
    int rowBase = wave * 16;

    // A: lane holds rows (rowBase+t), K pair {kk, kk+1}, kk = 4*step + 2*half
    const float* xrow = Xp + (size_t)(rowBase + t) * PD + 2 * half;
    const v2f*   bp   = Bpk + lane;

    v8f acc[CTILES];
#pragma unroll
    for (int c = 0; c < CTILES; ++c)
        acc[c] = (v8f){0.f, 0.f, 0.f, 0.f, 0.f, 0.f, 0.f, 0.f};

    for (int step = 0; step < KSTEPS; ++step) {
        v2f a = *(const v2f*)(xrow + step * 4);
#pragma unroll
        for (int c = 0; c < CTILES; ++c) {
            v2f b = bp[(step * CTILES + c) * 32];
            acc[c] = __builtin_amdgcn_wmma_f32_16x16x4_f32(
                false, a, false, b, (short)0, acc[c], false, false);
        }
    }

    // C/D layout: VGPR r -> (M = rowBase + r + 8*half, N = c*16 + t)
#pragma unroll
    for (int c = 0; c < CTILES; ++c) {
        int col = c * 16 + t;
        if (col < PD) {
#pragma unroll
            for (int r = 0; r < 8; ++r) {
                int m = rowBase + r + 8 * half;
                Hp[(size_t)m * PD + col] = acc[c][r];
            }
        }
    }
}

// ---------------------------------------------------------------------------
// Per-node attention logits: alpha_s = h . a_src, alpha_d = h . a_dst
// ---------------------------------------------------------------------------
__global__ __launch_bounds__(256)
void gat_alpha(const float* __restrict__ Hp, const float* __restrict__ a_src,
               const float* __restrict__ a_dst, float* __restrict__ as,
               float* __restrict__ ad) {
    int n = blockIdx.x * blockDim.x + threadIdx.x;
    if (n >= N_NODES) return;
    const float* h = Hp + (size_t)n * PD;
    float s = 0.f, d = 0.f;
#pragma unroll 4
    for (int i = 0; i < D; ++i) {
        float v = h[i];
        s = fmaf(v, a_src[i], s);
        d = fmaf(v, a_dst[i], d);
    }
    as[n] = s;
    ad[n] = d;
}

// ---------------------------------------------------------------------------
// Init: m = -inf (ordered), z = 0, out[n, d] = bias[d] (0 in pad cols)
// ---------------------------------------------------------------------------
__global__ __launch_bounds__(256)
void gat_init(unsigned* __restrict__ m, float* __restrict__ z,
              float* __restrict__ out, int stride,
              const float* __restrict__ bias) {
    int i = blockIdx.x * blockDim.x + threadIdx.x;
    if (i < N_NODES) { m[i] = ORD_NEG_INF; z[i] = 0.f; }
    if (i < N_NODES * stride) {
        int d = i % stride;
        out[i] = (d < D) ? bias[d] : 0.f;
    }
}

// ---------------------------------------------------------------------------
// Pass 1: segment max of leaky_relu(alpha_s[src] + alpha_d[dst]) over dst
// ---------------------------------------------------------------------------
__global__ __launch_bounds__(256)
void gat_edge_max(const long long* __restrict__ ei,
                  const float* __restrict__ as, const float* __restrict__ ad,
                  unsigned* __restrict__ m) {
    int e = blockIdx.x * blockDim.x + threadIdx.x;
    if (e >= EA) return;
    int s, d; edge_sd(ei, e, s, d);
    float v = as[s] + ad[d];
    v = v > 0.f ? v : NEG_SLOPE * v;
    atomicMax(&m[d], f2ord(v));
}

// ---------------------------------------------------------------------------
// Pass 2: w = exp(e - m[dst]); z[dst] += w  (stash w per edge)
// ---------------------------------------------------------------------------
__global__ __launch_bounds__(256)
void gat_edge_exp(const long long* __restrict__ ei,
                  const float* __restrict__ as, const float* __restrict__ ad,
                  const unsigned* __restrict__ m, float* __restrict__ z,
                  float* __restrict__ w) {
    int e = blockIdx.x * blockDim.x + threadIdx.x;
    if (e >= EA) return;
    int s, d; edge_sd(ei, e, s, d);
    float v = as[s] + ad[d];
    v = v > 0.f ? v : NEG_SLOPE * v;
    float wv = __expf(v - ord2f(m[d]));
    w[e] = wv;
    atomicAdd(&z[d], wv);
}

// ---------------------------------------------------------------------------
// Pass 3: out[dst,:] += (w/z[dst]) * h[src,:]  — one wave per edge,
// lanes cover the 101 features (coalesced row reads, fp32 L2 atomics).
// ---------------------------------------------------------------------------
__global__ __launch_bounds__(256)
void gat_edge_agg(const long long* __restrict__ ei,
                  const float* __restrict__ Hp, const float* __restrict__ w,
                  const float* __restrict__ z, float* __restrict__ out,
                  int ostride) {
    int gw = (int)((blockIdx.x * blockDim.x + threadIdx.x) >> 5);
    if (gw >= EA) return;
    int lane = threadIdx.x & 31;
    int s, d; edge_sd(ei, gw, s, d);
    float coef = w[gw] / (z[d] + 1e-16f);
    const float* hs = Hp + (size_t)s * PD;
    float* od = out + (size_t)d * ostride;
#pragma unroll
    for (int f = lane; f < D; f += 32)
        atomicAdd(&od[f], coef * hs[f]);
}

// ---------------------------------------------------------------------------
// Orchestration
// ---------------------------------------------------------------------------
extern "C" void kernel_launch(void* const* d_in, const int* in_sizes, int n_in,
                              void* d_out, int out_size, void* d_ws, size_t ws_size,
                              hipStream_t stream) {
    const float*     x     = (const float*)d_in[0];
    const long long* ei    = (const long long*)d_in[1];   // int64 [2, E]
    const float*     Ws    = (const float*)d_in[2];       // [L, D, D]
    const float*     a_src = (const float*)d_in[3];       // [L, D]
    const float*     a_dst = (const float*)d_in[4];       // [L, D]
    const float*     bias  = (const float*)d_in[5];       // [L, D]
    float*           out   = (float*)d_out;

    // Workspace carve-out (~67 MB total)
    char*  ws  = (char*)d_ws;
    size_t off = 0;
    auto carve = [&](size_t bytes) -> void* {
        void* p = ws + off;
        off += (bytes + 255) & ~(size_t)255;
        return p;
    };
    float*    Xp   = (float*)   carve((size_t)N_NODES * PD * sizeof(float));
    float*    Hp   = (float*)   carve((size_t)N_NODES * PD * sizeof(float));
    float*    bufB = (float*)   carve((size_t)N_NODES * PD * sizeof(float));
    v2f*      Bpk  = (v2f*)     carve((size_t)BPK_ELEMS * sizeof(v2f));
    float*    as   = (float*)   carve((size_t)N_NODES * sizeof(float));
    float*    ad   = (float*)   carve((size_t)N_NODES * sizeof(float));
    unsigned* m    = (unsigned*)carve((size_t)N_NODES * sizeof(unsigned));
    float*    z    = (float*)   carve((size_t)N_NODES * sizeof(float));
    float*    wE   = (float*)   carve((size_t)EA * sizeof(float));
    (void)ws_size; (void)in_sizes; (void)n_in; (void)out_size;

    const int TB = 256;
    int gPackX = (N_NODES * PD + TB - 1) / TB;
    int gPackB = (BPK_ELEMS + TB - 1) / TB;
    int gGemm  = (ROWTILES * 32 + TB - 1) / TB;
    int gNode  = (N_NODES + TB - 1) / TB;
    int gEdge  = (EA + TB - 1) / TB;
    int gAgg   = (int)(((size_t)EA * 32 + TB - 1) / TB);

    gat_pack_x<<<gPackX, TB, 0, stream>>>(x, Xp);

    const float* cur = Xp;               // stride-PD, zero-padded
    float* nxt[L_LAYERS]    = { bufB, out };
    int    nstride[L_LAYERS] = { PD, D };

    for (int l = 0; l < L_LAYERS; ++l) {
        const float* Wl = Ws    + (size_t)l * D * D;
        const float* sl = a_src + (size_t)l * D;
        const float* dl = a_dst + (size_t)l * D;
        const float* bl = bias  + (size_t)l * D;
        int gInit = (N_NODES * nstride[l] + TB - 1) / TB;

        gat_pack_b  <<<gPackB, TB, 0, stream>>>(Wl, Bpk);
        gat_gemm_wmma<<<gGemm, TB, 0, stream>>>(cur, Bpk, Hp);
        gat_alpha   <<<gNode, TB, 0, stream>>>(Hp, sl, dl, as, ad);
        gat_init    <<<gInit, TB, 0, stream>>>(m, z, nxt[l], nstride[l], bl);
        gat_edge_max<<<gEdge, TB, 0, stream>>>(ei, as, ad, m);
        gat_edge_exp<<<gEdge, TB, 0, stream>>>(ei, as, ad, m, z, wE);
        gat_edge_agg<<<gAgg,  TB, 0, stream>>>(ei, Hp, wE, z, nxt[l], nstride[l]);

        cur = nxt[l];
    }
}